// GAT_3994319585691
// MI455X (gfx1250) — compile-verified
//
#include <hip/hip_runtime.h>
#include <hip/hip_bf16.h>

typedef __attribute__((ext_vector_type(2))) float v2f;
typedef __attribute__((ext_vector_type(8))) float v8f;

#define NEG_SLOPE 0.2f

// ---------------------------------------------------------------------------
// Dense GEMM  H[N,64] = X[N,K] @ W[64,K]^T  using V_WMMA_F32_16X16X4_F32.
// Block = 128 threads = 4 waves. Each block does 16 rows; wave w does cols
// [16w, 16w+16). K-loop in steps of 4.
//
// f32 16x16x4 WMMA operand striping (ISA 7.12.2):
//   A (16x4, MxK): lanes 0-15 row M=lane, VGPR0=K0 VGPR1=K1;
//                  lanes 16-31 row M=lane-16, VGPR0=K2 VGPR1=K3.
//   B (4x16, KxN): mirrored with N in place of M.
//   C/D (16x16):   VGPR v -> (M = v + 8*(lane>=16), N = lane&15).
// N is a multiple of 16 here (50000 = 3125*16), so no row guard is needed and
// EXEC stays all-ones around the WMMA as required.
// ---------------------------------------------------------------------------
__global__ __launch_bounds__(128) void wmma_gemm_kernel(
    const float* __restrict__ X, const float* __restrict__ W,
    float* __restrict__ H, int K) {
  const int wave = threadIdx.x >> 5;
  const int lane = threadIdx.x & 31;
  const int half = lane >> 4;      // 0: K0/K1 , 1: K2/K3
  const int l15  = lane & 15;
  const int rowBase = blockIdx.x * 16;
  const int colBase = wave * 16;

  const float* xrow = X + (size_t)(rowBase + l15) * K + half * 2;
  const float* wrow = W + (size_t)(colBase + l15) * K + half * 2;

  v8f c = {};
  for (int k = 0; k < K; k += 4) {
    v2f a = *(const v2f*)(xrow + k);   // 8B-aligned (k%4==0, +half*2)
    v2f b = *(const v2f*)(wrow + k);
    c = __builtin_amdgcn_wmma_f32_16x16x4_f32(
        /*neg_a=*/false, a, /*neg_b=*/false, b,
        /*c_mod=*/(short)0, c, /*reuse_a=*/false, /*reuse_b=*/false);
  }

  float* hout = H + (size_t)(rowBase + half * 8) * 64 + colBase + l15;
#pragma unroll
  for (int v = 0; v < 8; ++v) hout[(size_t)v * 64] = c[v];
}

// alpha_s[i] = <h_i, a_src>, alpha_d[i] = <h_i, a_dst>
__global__ void alpha_kernel(const float* __restrict__ h,
                             const float* __restrict__ a_src,
                             const float* __restrict__ a_dst,
                             float* __restrict__ as, float* __restrict__ ad,
                             int N) {
  int i = blockIdx.x * blockDim.x + threadIdx.x;
  if (i >= N) return;
  const float* r = h + (size_t)i * 64;
  float s = 0.f, d = 0.f;
#pragma unroll 8
  for (int c = 0; c < 64; ++c) {
    float v = r[c];
    s += v * a_src[c];
    d += v * a_dst[c];
  }
  as[i] = s;
  ad[i] = d;
}

__global__ void init_md_kernel(float* __restrict__ m, float* __restrict__ denom,
                               int N) {
  int i = blockIdx.x * blockDim.x + threadIdx.x;
  if (i >= N) return;
  m[i] = -INFINITY;
  denom[i] = 0.f;
}

__global__ void zero_kernel(float* __restrict__ p, long long total) {
  long long i = (long long)blockIdx.x * blockDim.x + threadIdx.x;
  if (i >= total) return;
  p[i] = 0.f;
}

// Float atomic max via sign-split integer atomics (works for mixed signs).
__device__ __forceinline__ void atomicMaxF(float* addr, float v) {
  if (v >= 0.f)
    atomicMax((int*)addr, __float_as_int(v));
  else
    atomicMin((unsigned int*)addr, __float_as_uint(v));
}

__device__ __forceinline__ void edge_sd(const long long* __restrict__ ei,
                                        long long e, long long E,
                                        long long& s, long long& d) {
  if (e < E) {
    s = ei[e];
    d = ei[E + e];
  } else {          // appended self-loop (i, i)
    s = e - E;
    d = e - E;
  }
}

// e = leakyrelu(alpha_s[src] + alpha_d[dst]); segment-max into m[dst]
__global__ void logits_kernel(const float* __restrict__ as,
                              const float* __restrict__ ad,
                              const long long* __restrict__ ei, long long E,
                              int N, float* __restrict__ elog,
                              float* __restrict__ m) {
  long long e = (long long)blockIdx.x * blockDim.x + threadIdx.x;
  if (e >= E + N) return;
  long long s, d;
  edge_sd(ei, e, E, s, d);
  float v = as[s] + ad[d];
  v = v > 0.f ? v : NEG_SLOPE * v;
  elog[e] = v;
  atomicMaxF(m + d, v);
}

// ex = exp(e - m[dst]); segment-sum into denom[dst]
__global__ void expsum_kernel(const float* __restrict__ elog,
                              const float* __restrict__ m,
                              const long long* __restrict__ ei, long long E,
                              int N, float* __restrict__ eexp,
                              float* __restrict__ denom) {
  long long e = (long long)blockIdx.x * blockDim.x + threadIdx.x;
  if (e >= E + N) return;
  long long s, d;
  edge_sd(ei, e, E, s, d);
  float ex = __expf(elog[e] - m[d]);
  eexp[e] = ex;
  atomicAdd(denom + d, ex);
}

// agg[dst] += (ex/denom[dst]) * h[src]   — one wave (32 lanes) per edge,
// lane covers channels lane and lane+32: two coalesced 128B lines per edge.
__global__ __launch_bounds__(256) void agg_kernel(
    const float* __restrict__ eexp, const float* __restrict__ denom,
    const float* __restrict__ h, const long long* __restrict__ ei, long long E,
    int N, float* __restrict__ agg) {
  const int lane = threadIdx.x & 31;
  long long e = (long long)blockIdx.x * 8 + (threadIdx.x >> 5);
  if (e >= E + N) return;
  long long s, d;
  edge_sd(ei, e, E, s, d);
  float coef = eexp[e] / denom[d];
  const float* hs = h + s * 64;
  float* od = agg + d * 64;
  atomicAdd(od + lane,      coef * hs[lane]);
  atomicAdd(od + lane + 32, coef * hs[lane + 32]);
}

// out = agg + bias, optional ReLU
__global__ void bias_act_kernel(const float* __restrict__ agg,
                                const float* __restrict__ b,
                                float* __restrict__ out, int do_relu,
                                long long total) {
  long long i = (long long)blockIdx.x * blockDim.x + threadIdx.x;
  if (i >= total) return;
  float v = agg[i] + b[i & 63];
  out[i] = do_relu ? fmaxf(v, 0.f) : v;
}

extern "C" void kernel_launch(void* const* d_in, const int* in_sizes, int n_in,
                              void* d_out, int out_size, void* d_ws,
                              size_t ws_size, hipStream_t stream) {
  const float* x        = (const float*)d_in[0];
  const long long* ei   = (const long long*)d_in[1];
  const float* W1       = (const float*)d_in[2];
  const float* a1_src   = (const float*)d_in[3];
  const float* a1_dst   = (const float*)d_in[4];
  const float* b1       = (const float*)d_in[5];
  const float* W2       = (const float*)d_in[6];
  const float* a2_src   = (const float*)d_in[7];
  const float* a2_dst   = (const float*)d_in[8];
  const float* b2       = (const float*)d_in[9];
  float* out            = (float*)d_out;

  const int N = in_sizes[0] / 128;               // 50000
  const long long E = (long long)in_sizes[1] / 2; // 800000
  const long long Etot = E + N;

  // workspace layout (floats)
  float* ws      = (float*)d_ws;
  float* hbuf    = ws;                    // N*64
  float* aggbuf  = hbuf + (size_t)N * 64; // N*64
  float* alpha_s = aggbuf + (size_t)N * 64;
  float* alpha_d = alpha_s + N;
  float* mbuf    = alpha_d + N;
  float* denom   = mbuf + N;
  float* elog    = denom + N;             // Etot
  float* eexp    = elog + Etot;           // Etot

  const int TB = 256;
  const int nodeBlocks  = (N + TB - 1) / TB;
  const int edgeBlocks  = (int)((Etot + TB - 1) / TB);
  const int aggBlocks   = (int)((Etot + 7) / 8);   // 8 waves (edges) per block
  const long long feat  = (long long)N * 64;
  const int featBlocks  = (int)((feat + TB - 1) / TB);
  const int gemmBlocks  = N / 16;                  // 3125

  // ---------------- layer 1 ----------------
  wmma_gemm_kernel<<<gemmBlocks, 128, 0, stream>>>(x, W1, hbuf, 128);
  alpha_kernel<<<nodeBlocks, TB, 0, stream>>>(hbuf, a1_src, a1_dst, alpha_s,
                                              alpha_d, N);
  init_md_kernel<<<nodeBlocks, TB, 0, stream>>>(mbuf, denom, N);
  logits_kernel<<<edgeBlocks, TB, 0, stream>>>(alpha_s, alpha_d, ei, E, N,
                                               elog, mbuf);
  expsum_kernel<<<edgeBlocks, TB, 0, stream>>>(elog, mbuf, ei, E, N, eexp,
                                               denom);
  zero_kernel<<<featBlocks, TB, 0, stream>>>(aggbuf, feat);
  agg_kernel<<<aggBlocks, TB, 0, stream>>>(eexp, denom, hbuf, ei, E, N,
                                           aggbuf);
  // h_relu = relu(agg + b1) -> overwrite hbuf
  bias_act_kernel<<<featBlocks, TB, 0, stream>>>(aggbuf, b1, hbuf, 1, feat);

  // ---------------- layer 2 ----------------
  wmma_gemm_kernel<<<gemmBlocks, 128, 0, stream>>>(hbuf, W2, aggbuf, 64);
  alpha_kernel<<<nodeBlocks, TB, 0, stream>>>(aggbuf, a2_src, a2_dst, alpha_s,
                                              alpha_d, N);
  init_md_kernel<<<nodeBlocks, TB, 0, stream>>>(mbuf, denom, N);
  logits_kernel<<<edgeBlocks, TB, 0, stream>>>(alpha_s, alpha_d, ei, E, N,
                                               elog, mbuf);
  expsum_kernel<<<edgeBlocks, TB, 0, stream>>>(elog, mbuf, ei, E, N, eexp,
                                               denom);
  zero_kernel<<<featBlocks, TB, 0, stream>>>(hbuf, feat);   // hbuf free now
  agg_kernel<<<aggBlocks, TB, 0, stream>>>(eexp, denom, aggbuf, ei, E, N,
                                           hbuf);
  bias_act_kernel<<<featBlocks, TB, 0, stream>>>(hbuf, b2, out, 0, feat);
}